// SelectiveStateSpaceModel_51969104281933
// MI455X (gfx1250) — compile-verified
//
#include <hip/hip_runtime.h>

// Mamba selective scan for MI455X (gfx1250, wave32).
//   L=2048, d_inner=4096, dt_rank=128, d_state=16
// Phase 1: dbc = x @ W_in^T           fp32 WMMA 16x16x4, K split 4-ways for
//                                     latency coverage, 4 acc chains/wave.
// Phase 1b: reduce K-partials (fixed order -> deterministic).
// Phase 2: delta = softplus(dbc[:, :128] @ W_delta^T) -> written into d_out.
// Phase 3: sequential scan, 1 thread per (d,n); y overwrites delta in d_out.

#define SEQ_L   2048
#define DI      4096
#define RTOT    160     // dt_rank + 2*d_state
#define DTR     128
#define DS      16
#define KSPLIT  4
#define KCH     (DI / KSPLIT)        // 1024
#define DBC_N   (SEQ_L * RTOT)       // 327680 elements per dbc image

typedef float v2f __attribute__((ext_vector_type(2)));
typedef float v8f __attribute__((ext_vector_type(8)));

static __device__ __forceinline__ v8f wmma4(v2f a, v2f b, v8f c) {
    return __builtin_amdgcn_wmma_f32_16x16x4_f32(
               false, a, false, b, (short)0, c, false, false);
}

// ---------------------------------------------------------------------------
// GEMM1: part[ks][L,160] = x[:, ks*1024:(ks+1)*1024] @ W_in[:, same]^T
// One wave = one 16x16 tile of one K-chunk. 1280 tiles x 4 chunks = 5120 waves.
// 4 independent WMMA accumulation chains; 32 b64 loads in flight per wave.
// ---------------------------------------------------------------------------
__global__ void __launch_bounds__(128)
sssm_gemm_in(const float* __restrict__ x, const float* __restrict__ Win,
             float* __restrict__ part) {
    const int wave = (blockIdx.x * 128 + threadIdx.x) >> 5;
    const int lane = threadIdx.x & 31;
    const int ks   = wave & (KSPLIT - 1);          // K chunk
    const int tile = wave >> 2;                    // 0..1279
    const int l0 = (tile / (RTOT / 16)) * 16;
    const int j0 = (tile % (RTOT / 16)) * 16;

    const int m  = lane & 15;          // A row / B col held by this lane
    const int kh = (lane >> 4) << 1;   // K sub-offset: 0 (lanes 0-15) or 2 (16-31)

    const float* ap = x   + (size_t)(l0 + m) * DI + ks * KCH + kh;
    const float* bp = Win + (size_t)(j0 + m) * DI + ks * KCH + kh;

    v8f acc0 = {}, acc1 = {}, acc2 = {}, acc3 = {};
    #pragma unroll 4
    for (int k = 0; k < KCH; k += 16) {
        v2f a0 = *(const v2f*)(ap + 0),  b0 = *(const v2f*)(bp + 0);
        v2f a1 = *(const v2f*)(ap + 4),  b1 = *(const v2f*)(bp + 4);
        v2f a2 = *(const v2f*)(ap + 8),  b2 = *(const v2f*)(bp + 8);
        v2f a3 = *(const v2f*)(ap + 12), b3 = *(const v2f*)(bp + 12);
        acc0 = wmma4(a0, b0, acc0);
        acc1 = wmma4(a1, b1, acc1);
        acc2 = wmma4(a2, b2, acc2);
        acc3 = wmma4(a3, b3, acc3);
        ap += 16; bp += 16;
    }
    v8f acc = (acc0 + acc1) + (acc2 + acc3);

    const int mrow = (lane >> 4) << 3;             // +8 for upper half-wave
    float* dst = part + (size_t)ks * DBC_N;
    #pragma unroll
    for (int r = 0; r < 8; ++r)
        dst[(size_t)(l0 + mrow + r) * RTOT + j0 + (lane & 15)] = acc[r];
}

// ---------------------------------------------------------------------------
// Reduce the 4 K-partials into dbc (fixed summation order -> deterministic).
// ---------------------------------------------------------------------------
__global__ void __launch_bounds__(256)
sssm_reduce(const float* __restrict__ part, float* __restrict__ dbc) {
    const int i = blockIdx.x * 256 + threadIdx.x;
    dbc[i] = (part[i] + part[i + DBC_N]) +
             (part[i + 2 * (size_t)DBC_N] + part[i + 3 * (size_t)DBC_N]);
}

// ---------------------------------------------------------------------------
// GEMM2: delta[L,4096] = softplus(dbc[:, 0:128] @ W_delta[4096,128]^T)
// One wave per 16x16 tile, K=128 -> 32 WMMAs in 4 chains. 32768 waves.
// Result written into d_out (consumed in-place by the scan).
// ---------------------------------------------------------------------------
__global__ void __launch_bounds__(256)
sssm_gemm_delta(const float* __restrict__ dbc, const float* __restrict__ Wd,
                float* __restrict__ delta) {
    const int wave = (blockIdx.x * 256 + threadIdx.x) >> 5;
    const int lane = threadIdx.x & 31;
    const int l0 = (wave >> 8) * 16;       // 256 column tiles (4096/16)
    const int d0 = (wave & 255) * 16;

    const int m  = lane & 15;
    const int kh = (lane >> 4) << 1;

    const float* ap = dbc + (size_t)(l0 + m) * RTOT + kh;  // delta_raw cols 0..127
    const float* bp = Wd  + (size_t)(d0 + m) * DTR  + kh;

    v8f acc0 = {}, acc1 = {}, acc2 = {}, acc3 = {};
    #pragma unroll
    for (int k = 0; k < DTR; k += 16) {
        v2f a0 = *(const v2f*)(ap + 0),  b0 = *(const v2f*)(bp + 0);
        v2f a1 = *(const v2f*)(ap + 4),  b1 = *(const v2f*)(bp + 4);
        v2f a2 = *(const v2f*)(ap + 8),  b2 = *(const v2f*)(bp + 8);
        v2f a3 = *(const v2f*)(ap + 12), b3 = *(const v2f*)(bp + 12);
        acc0 = wmma4(a0, b0, acc0);
        acc1 = wmma4(a1, b1, acc1);
        acc2 = wmma4(a2, b2, acc2);
        acc3 = wmma4(a3, b3, acc3);
        ap += 16; bp += 16;
    }
    v8f acc = (acc0 + acc1) + (acc2 + acc3);

    const int mrow = (lane >> 4) << 3;
    #pragma unroll
    for (int r = 0; r < 8; ++r) {
        float v  = acc[r];
        // numerically stable softplus: max(v,0) + log1p(exp(-|v|))
        float sp = fmaxf(v, 0.0f) + log1pf(__expf(-fabsf(v)));
        delta[(size_t)(l0 + mrow + r) * DI + d0 + (lane & 15)] = sp;
    }
}

// ---------------------------------------------------------------------------
// Scan: one thread per (d, n). 65536 threads = 2048 wave32s.
// h kept in a register; y = sum_n h*C reduced via 4 shfl_xor steps inside the
// 16-lane group. delta lives in `out` and is overwritten with y by lane n==0
// (the same wave reads it earlier in program order -> no race).
// Unroll x4 so loads for future steps issue ahead of the exp->fma chain.
// ---------------------------------------------------------------------------
__global__ void __launch_bounds__(256)
sssm_scan(const float* __restrict__ x, const float* __restrict__ A_log,
          const float* __restrict__ Dvec, const float* __restrict__ dbc,
          float* __restrict__ out) {
    const int t = blockIdx.x * 256 + threadIdx.x;
    const int d = t >> 4;
    const int n = t & 15;

    const float A  = -__expf(A_log[(size_t)d * DS + n]);   // A = -exp(A_log)
    const float Dd = Dvec[d];

    float h = 0.0f;
    const float* xp  = x   + d;
    const float* dp  = out + d;              // delta stream (aliased with output)
    const float* bcp = dbc + DTR + n;        // B at col 128+n, C at 144+n

    #pragma unroll 4
    for (int l = 0; l < SEQ_L; ++l) {
        float dl = dp[(size_t)l * DI];
        float xl = xp[(size_t)l * DI];
        float Bl = bcp[(size_t)l * RTOT];
        float Cl = bcp[(size_t)l * RTOT + DS];

        float dA = __expf(dl * A);           // v_exp_f32 (critical path)
        h = fmaf(dA, h, dl * Bl * xl);       // h = dA*h + delta*B*x
        float y = h * Cl;

        // reduce over the 16 n-lanes (xor masks stay inside the group)
        y += __shfl_xor(y, 1, 32);
        y += __shfl_xor(y, 2, 32);
        y += __shfl_xor(y, 4, 32);
        y += __shfl_xor(y, 8, 32);

        if (n == 0)
            out[(size_t)l * DI + d] = fmaf(xl, Dd, y);
    }
}

// ---------------------------------------------------------------------------
extern "C" void kernel_launch(void* const* d_in, const int* in_sizes, int n_in,
                              void* d_out, int out_size, void* d_ws, size_t ws_size,
                              hipStream_t stream) {
    const float* x    = (const float*)d_in[0];   // [2048, 4096]
    const float* Win  = (const float*)d_in[1];   // [160, 4096]
    const float* Wd   = (const float*)d_in[2];   // [4096, 128]
    const float* Alog = (const float*)d_in[3];   // [4096, 16]
    const float* Dv   = (const float*)d_in[4];   // [4096]
    float*       out  = (float*)d_out;           // [2048, 4096]

    float* part = (float*)d_ws;                  // [4][2048,160] = 5.24 MB
    float* dbc  = part + (size_t)KSPLIT * DBC_N; // [2048,160]    = 1.31 MB

    // GEMM1 partials: 5120 waves, 4 waves/block
    sssm_gemm_in<<<1280, 128, 0, stream>>>(x, Win, part);
    // K-partial reduction: 327680 elements
    sssm_reduce<<<DBC_N / 256, 256, 0, stream>>>(part, dbc);
    // GEMM2 (+softplus) into d_out: 32768 waves, 8 waves/block
    sssm_gemm_delta<<<4096, 256, 0, stream>>>(dbc, Wd, out);
    // Scan: 65536 threads
    sssm_scan<<<(DI * DS) / 256, 256, 0, stream>>>(x, Alog, Dv, dbc, out);
}